// LaplacianPositionalEncoding_12429635354775
// MI455X (gfx1250) — compile-verified
//
#include <hip/hip_runtime.h>
#include <stdint.h>
#include <math.h>

// Problem constants (match reference)
#define G_      1024
#define N_      128
#define F_      256
#define DIM_    16
#define EPG_    1024      // directed edges per graph, contiguous per graph
#define LDA_    129       // padded LDS row stride: (129*i+p) % 64 = (i+p) % 64 -> conflict-free column walks
#define NSWEEPS_ 10

typedef __attribute__((ext_vector_type(2))) float v2f;
typedef __attribute__((ext_vector_type(8))) float v8f;

// LDS layout (floats then ints):
//   A   : 128*129
//   V   : 128*129
//   Ksh : 16*256       (projection kernel, async-loaded)
//   Pos : 128*16       (selected eigenvectors)
//   cs  : 128          (c[0..63], s[64..127])
//   eval: 128
//   ints: degi[128] colOf[16]  (=144)
// total = 158,272 B -> 2 workgroups fit in a 320KB WGP
static constexpr size_t LDS_FLOATS = (size_t)(2 * N_ * LDA_ + DIM_ * F_ + N_ * DIM_ + 128 + 128);
static constexpr size_t LDS_BYTES  = (LDS_FLOATS + 144) * 4;

// Closed-form round-robin (circle method) pairing: n=128 players, player 127 fixed.
// Round r in [0,126], pair k in [0,63]; covers all C(128,2) pairs over 127 rounds,
// pairs within a round are disjoint.
__device__ __forceinline__ void rr_pair(int r, int k, int& p, int& q) {
    int a, b;
    if (k == 0) {
        a = 127;
        b = r % 127;
    } else {
        a = (r + k) % 127;
        b = (r - k + 127) % 127;
    }
    p = min(a, b);
    q = max(a, b);
}

extern "C" __global__ __launch_bounds__(256, 2)
void lpe_fused_kernel(const float* __restrict__ nf, const float* __restrict__ Kmat,
                      const int* __restrict__ esrc, const int* __restrict__ edst,
                      float* __restrict__ out)
{
    extern __shared__ float smem[];
    float* A    = smem;
    float* V    = A + N_ * LDA_;
    float* Ksh  = V + N_ * LDA_;
    float* Pos  = Ksh + DIM_ * F_;
    float* cs   = Pos + N_ * DIM_;
    float* eval = cs + 128;
    int*   degi = (int*)(eval + 128);
    int*   colOf= degi + 128;

    const int tid  = threadIdx.x;
    const int g    = blockIdx.x;
    const int wid  = tid >> 5;     // wave32
    const int lane = tid & 31;

    // ---- async-load the 16x256 f32 projection kernel into LDS (ASYNCcnt),
    //      overlapped with the whole eigensolve ----
    {
        unsigned ldsBase = (unsigned)(uintptr_t)(void*)Ksh;   // low 32 bits = LDS offset
        unsigned long long gbase = (unsigned long long)(uintptr_t)Kmat;
        #pragma unroll
        for (int i = 0; i < 4; ++i) {
            int chunk = tid + 256 * i;                         // 1024 x 16B = 16KB
            unsigned loff = ldsBase + (unsigned)(chunk * 16);
            unsigned long long ga = gbase + (unsigned long long)(chunk * 16);
            asm volatile("global_load_async_to_lds_b128 %0, %1, off"
                         :: "v"(loff), "v"(ga) : "memory");
        }
    }

    // ---- L2 prefetch of this block's node-feature slice (128KB); total 128MB < 192MB L2 ----
    {
        const char* base = (const char*)(nf + (size_t)g * N_ * F_);
        #pragma unroll
        for (int i = 0; i < 4; ++i)
            __builtin_prefetch(base + (size_t)(tid + 256 * i) * 128, 0, 1);
    }

    // ---- degrees (bincount of edge_dst, local ids) ----
    if (tid < N_) degi[tid] = 0;
    __syncthreads();
    for (int e = tid; e < EPG_; e += 256) {
        int d = edst[g * EPG_ + e] & (N_ - 1);
        atomicAdd(&degi[d], 1);
    }
    __syncthreads();

    // ---- A = I, V = I ----
    for (int t = tid; t < N_ * N_; t += 256) {
        int i = t >> 7, j = t & 127;
        float v = (i == j) ? 1.0f : 0.0f;
        A[i * LDA_ + j] = v;
        V[i * LDA_ + j] = v;
    }
    __syncthreads();

    // ---- scatter -adj_norm into A (LDS float atomics; duplicate edges accumulate) ----
    for (int e = tid; e < EPG_; e += 256) {
        int s = esrc[g * EPG_ + e] & (N_ - 1);
        int d = edst[g * EPG_ + e] & (N_ - 1);
        float w = -rsqrtf((float)degi[s] * (float)degi[d]);
        atomicAdd(&A[s * LDA_ + d], w);
    }
    __syncthreads();

    // ---- parallel cyclic Jacobi: 64 disjoint rotations/step, 127 steps/sweep ----
    for (int sweep = 0; sweep < NSWEEPS_; ++sweep) {
        for (int step = 0; step < N_ - 1; ++step) {
            if (tid < 64) {
                int p, q;
                rr_pair(step, tid, p, q);
                float app = A[p * LDA_ + p];
                float aqq = A[q * LDA_ + q];
                float apq = A[p * LDA_ + q];
                float c = 1.0f, s = 0.0f;
                if (fabsf(apq) > 1e-12f) {
                    float tau = (aqq - app) / (2.0f * apq);
                    float tt  = copysignf(1.0f, tau) / (fabsf(tau) + sqrtf(1.0f + tau * tau));
                    c = rsqrtf(1.0f + tt * tt);
                    s = tt * c;
                }
                cs[tid] = c; cs[64 + tid] = s;
            }
            __syncthreads();
            // column phase: A <- A*J, V <- V*J (pairs write disjoint columns)
            for (int t = tid; t < 64 * N_; t += 256) {
                int k = t >> 7, i = t & 127;
                int p, q;
                rr_pair(step, k, p, q);
                float c = cs[k], s = cs[64 + k];
                float x = A[i * LDA_ + p], y = A[i * LDA_ + q];
                A[i * LDA_ + p] = c * x - s * y;
                A[i * LDA_ + q] = s * x + c * y;
                x = V[i * LDA_ + p]; y = V[i * LDA_ + q];
                V[i * LDA_ + p] = c * x - s * y;
                V[i * LDA_ + q] = s * x + c * y;
            }
            __syncthreads();
            // row phase: A <- J^T * A (pairs write disjoint rows)
            for (int t = tid; t < 64 * N_; t += 256) {
                int k = t >> 7, j = t & 127;
                int p, q;
                rr_pair(step, k, p, q);
                float c = cs[k], s = cs[64 + k];
                float x = A[p * LDA_ + j], y = A[q * LDA_ + j];
                A[p * LDA_ + j] = c * x - s * y;
                A[q * LDA_ + j] = s * x + c * y;
            }
            __syncthreads();
        }
    }

    // ---- eigenvalue ranking (ascending, ties by index), select ranks 1..16 ----
    if (tid < N_) eval[tid] = A[tid * LDA_ + tid];
    __syncthreads();
    if (tid < N_) {
        float vi = eval[tid];
        int r = 0;
        for (int j = 0; j < N_; ++j) {
            float vj = eval[j];
            r += (vj < vi) || (vj == vi && j < tid);
        }
        if (r >= 1 && r <= DIM_) colOf[r - 1] = tid;
    }
    __syncthreads();
    for (int t = tid; t < N_ * DIM_; t += 256) {
        int row = t >> 4, d = t & 15;
        Pos[row * DIM_ + d] = V[row * LDA_ + colOf[d]];
    }
    // Ksh async load must have landed before the WMMA stage reads it
    asm volatile("s_wait_asynccnt 0" ::: "memory");
    __syncthreads();

    // ---- out = node_feature + Pos[128x16] @ K[16x256], f32 WMMA 16x16x4, K chained x4 ----
    {
        const int m0   = wid * 16;        // 8 waves -> 8 row tiles of 16
        const int half = lane >> 4;       // 0: lanes 0-15, 1: lanes 16-31
        const int l15  = lane & 15;

        // A-matrix fragments (16x4 f32): vgpr0 = {K0,K2}, vgpr1 = {K1,K3} per half
        v2f a[4];
        {
            int row = m0 + l15;
            #pragma unroll
            for (int j2 = 0; j2 < 4; ++j2) {
                int k0 = 4 * j2 + 2 * half;
                a[j2].x = Pos[row * DIM_ + k0];
                a[j2].y = Pos[row * DIM_ + k0 + 1];
            }
        }

        const size_t rowBase = (size_t)g * N_ * F_;
        for (int nt = 0; nt < F_ / 16; ++nt) {
            int col = nt * 16 + l15;
            v2f b[4];
            #pragma unroll
            for (int j2 = 0; j2 < 4; ++j2) {
                int k0 = 4 * j2 + 2 * half;
                b[j2].x = Ksh[k0 * F_ + col];
                b[j2].y = Ksh[(k0 + 1) * F_ + col];
            }
            v8f c;
            #pragma unroll
            for (int r = 0; r < 8; ++r) {
                int row = m0 + r + 8 * half;
                c[r] = nf[rowBase + (size_t)row * F_ + col];   // accumulator starts at node_feature
            }
            c = __builtin_amdgcn_wmma_f32_16x16x4_f32(false, a[0], false, b[0], (short)0, c, false, false);
            c = __builtin_amdgcn_wmma_f32_16x16x4_f32(false, a[1], false, b[1], (short)0, c, false, false);
            c = __builtin_amdgcn_wmma_f32_16x16x4_f32(false, a[2], false, b[2], (short)0, c, false, false);
            c = __builtin_amdgcn_wmma_f32_16x16x4_f32(false, a[3], false, b[3], (short)0, c, false, false);
            #pragma unroll
            for (int r = 0; r < 8; ++r) {
                int row = m0 + r + 8 * half;
                out[rowBase + (size_t)row * F_ + col] = c[r];
            }
        }
    }
}

extern "C" void kernel_launch(void* const* d_in, const int* in_sizes, int n_in,
                              void* d_out, int out_size, void* d_ws, size_t ws_size,
                              hipStream_t stream) {
    (void)in_sizes; (void)n_in; (void)out_size; (void)d_ws; (void)ws_size;
    const float* nf   = (const float*)d_in[0];
    const float* Kmat = (const float*)d_in[1];
    const int*   esrc = (const int*)d_in[2];
    const int*   edst = (const int*)d_in[3];
    float* out = (float*)d_out;

    // allow >64KB dynamic LDS (CDNA5 WGP: 320KB)
    hipFuncSetAttribute((const void*)lpe_fused_kernel,
                        hipFuncAttributeMaxDynamicSharedMemorySize, (int)LDS_BYTES);

    lpe_fused_kernel<<<G_, 256, LDS_BYTES, stream>>>(nf, Kmat, esrc, edst, out);
}